// GraphAttentionLayer_30958124270123
// MI455X (gfx1250) — compile-verified
//
#include <hip/hip_runtime.h>

typedef float v2f __attribute__((ext_vector_type(2)));
typedef float v8f __attribute__((ext_vector_type(8)));

#define N_NODES 4096
#define C_IN    512
#define N_HEADS 8
#define N_HIDDEN 64
#define OUTC    512   // N_HEADS * N_HIDDEN

static __device__ __forceinline__ v8f wmma_f32(v2f a, v2f b, v8f c) {
  // V_WMMA_F32_16X16X4_F32: (neg_a, A, neg_b, B, c_mod, C, reuse_a, reuse_b)
  return __builtin_amdgcn_wmma_f32_16x16x4_f32(false, a, false, b, (short)0, c,
                                               false, false);
}

// ---------------------------------------------------------------------------
// K1: g = h @ W^T   (4096x512) = (4096x512)(512x512)
// Register-blocked: one wave -> 16 rows x 64 cols (4 N-tiles sharing A frag).
// A frag: lane<16 -> h[row=lane][k,k+1], lane>=16 -> [k+2,k+3]       (b64)
// B frag: B[k][n] = W[n][k]; W row-major => (k',k'+1) contiguous     (b64)
// ---------------------------------------------------------------------------
__global__ void gat_k1_gemm(const float* __restrict__ h, const float* __restrict__ w,
                            float* __restrict__ g) {
  const int lane = threadIdx.x & 31;
  const int wave = threadIdx.x >> 5;
  const int tile = blockIdx.x * 8 + wave;        // 2048 = 256 it * 8 groups
  const int it = tile >> 3;
  const int gr = tile & 7;                       // 64-col group
  const int r    = lane & 15;
  const int half = lane >> 4;

  const float* ap = h + (size_t)(it * 16 + r) * C_IN + 2 * half;
  const float* bp[4];
#pragma unroll
  for (int t = 0; t < 4; ++t)
    bp[t] = w + (size_t)(gr * 64 + t * 16 + r) * C_IN + 2 * half;

  v8f acc[4];
#pragma unroll
  for (int t = 0; t < 4; ++t) acc[t] = (v8f){0.f,0.f,0.f,0.f,0.f,0.f,0.f,0.f};

#pragma unroll 2
  for (int k = 0; k < C_IN; k += 4) {
    v2f a = *(const v2f*)(ap + k);
#pragma unroll
    for (int t = 0; t < 4; ++t) {
      v2f b = *(const v2f*)(bp[t] + k);
      acc[t] = wmma_f32(a, b, acc[t]);
    }
  }
#pragma unroll
  for (int t = 0; t < 4; ++t) {
    float* op = g + (size_t)it * 16 * OUTC + gr * 64 + t * 16 + r;
#pragma unroll
    for (int vr = 0; vr < 8; ++vr)
      op[(size_t)(vr + 8 * half) * OUTC] = acc[t][vr];
  }
}

// ---------------------------------------------------------------------------
// K2a: e_j[n,h] = dot(g[n,h,:], a_r)   (a_r = attn_weight[64..127])
// ---------------------------------------------------------------------------
__global__ void gat_k2a_ej(const float* __restrict__ g, const float* __restrict__ aw,
                           float* __restrict__ ej) {
  const int id = blockIdx.x * 256 + threadIdx.x;   // 0..32767
  const int n  = id >> 3;
  const int hd = id & 7;
  const float* gp = g + (size_t)n * OUTC + hd * N_HIDDEN;
  const float* ar = aw + N_HIDDEN;
  float s = 0.f;
#pragma unroll 8
  for (int f = 0; f < N_HIDDEN; ++f) s += gp[f] * ar[f];
  ej[id] = s;
}

// ---------------------------------------------------------------------------
// K2b: per-head global max M[h] = max_n e_j[n,h]  (single block)
// ---------------------------------------------------------------------------
__global__ void gat_k2b_max(const float* __restrict__ ej, float* __restrict__ M) {
  __shared__ float red[256];
  const int t = threadIdx.x;
  const int hd = t & 7, slot = t >> 3;
  float m = -3.4e38f;
  for (int n = slot; n < N_NODES; n += 32) m = fmaxf(m, ej[n * 8 + hd]);
  red[hd * 32 + slot] = m;
  __syncthreads();
  if (t < 8) {
    float mm = red[t * 32];
    for (int s = 1; s < 32; ++s) mm = fmaxf(mm, red[t * 32 + s]);
    M[t] = mm;
  }
}

// ---------------------------------------------------------------------------
// K2t: LDS-tiled transpose + scale:  bt[c][n] = E[n, c>>6] * g[n][c]
//      bt is 512 x 4096 so K3's B fragments become contiguous b64 loads.
// ---------------------------------------------------------------------------
__global__ void gat_k2t_transpose(const float* __restrict__ g,
                                  const float* __restrict__ ej,
                                  const float* __restrict__ M,
                                  float* __restrict__ bt) {
  __shared__ float tile[32][33];
  const int c0 = blockIdx.x * 32;                // 0..511  (16 blocks)
  const int n0 = blockIdx.y * 32;                // 0..4095 (128 blocks)
  const int tx = threadIdx.x & 31;
  const int ty = threadIdx.x >> 5;               // 32x8 threads
  const int hd = (c0 + tx) >> 6;
  const float Mh = M[hd];
#pragma unroll
  for (int i = 0; i < 32; i += 8) {
    const int n = n0 + ty + i;
    const float E = __expf(ej[n * 8 + hd] - Mh);
    tile[ty + i][tx] = g[(size_t)n * OUTC + c0 + tx] * E;
  }
  __syncthreads();
#pragma unroll
  for (int i = 0; i < 32; i += 8) {
    const int c = c0 + ty + i;
    bt[(size_t)c * N_NODES + n0 + tx] = tile[tx][ty + i];
  }
}

// ---------------------------------------------------------------------------
// K2e: eextT[c][n] = (c<8) ? E[n,c] : 0     (16 x 4096, denominator B block)
// ---------------------------------------------------------------------------
__global__ void gat_k2e_eext(const float* __restrict__ ej, const float* __restrict__ M,
                             float* __restrict__ eextT) {
  const int id = blockIdx.x * 256 + threadIdx.x;  // 0..16*4096-1
  const int c = id >> 12;
  const int n = id & 4095;
  eextT[id] = (c < 8) ? __expf(ej[n * 8 + c] - M[c]) : 0.f;
}

// ---------------------------------------------------------------------------
// K3: [num | den] = Adj (4096x4096, {0,1}) x B (4096x528)
//     Register-blocked 16x64 strips; B read transposed (bt / eextT) => b64.
//     gr 0..7: 4 tiles into d_out;  gr 8: 1 tile of denominators.
// ---------------------------------------------------------------------------
__global__ void gat_k3_attn(const int* __restrict__ adj, const float* __restrict__ bt,
                            const float* __restrict__ eextT, float* __restrict__ out,
                            float* __restrict__ den) {
  const int lane = threadIdx.x & 31;
  const int wave = threadIdx.x >> 5;
  const int tile = blockIdx.x * 8 + wave;         // 2304 = 256 it * 9 groups
  const int it = tile / 9;
  const int gr = tile - it * 9;
  const int r    = lane & 15;
  const int half = lane >> 4;

  const int* ap = adj + (size_t)(it * 16 + r) * N_NODES + 2 * half;

  if (gr < 8) {
    const float* bp[4];
#pragma unroll
    for (int t = 0; t < 4; ++t)
      bp[t] = bt + (size_t)(gr * 64 + t * 16 + r) * N_NODES + 2 * half;

    v8f acc[4];
#pragma unroll
    for (int t = 0; t < 4; ++t) acc[t] = (v8f){0.f,0.f,0.f,0.f,0.f,0.f,0.f,0.f};

#pragma unroll 2
    for (int k = 0; k < N_NODES; k += 4) {
      int2 ai = *(const int2*)(ap + k);
      v2f a; a.x = (float)ai.x; a.y = (float)ai.y;   // exact: adj in {0,1}
#pragma unroll
      for (int t = 0; t < 4; ++t) {
        v2f b = *(const v2f*)(bp[t] + k);
        acc[t] = wmma_f32(a, b, acc[t]);
      }
    }
#pragma unroll
    for (int t = 0; t < 4; ++t) {
      float* op = out + (size_t)it * 16 * OUTC + gr * 64 + t * 16 + r;
#pragma unroll
      for (int vr = 0; vr < 8; ++vr)
        op[(size_t)(vr + 8 * half) * OUTC] = acc[t][vr];
    }
  } else {
    const float* bp = eextT + (size_t)r * N_NODES + 2 * half;
    v8f acc = {0.f,0.f,0.f,0.f,0.f,0.f,0.f,0.f};
#pragma unroll 4
    for (int k = 0; k < N_NODES; k += 4) {
      int2 ai = *(const int2*)(ap + k);
      v2f a; a.x = (float)ai.x; a.y = (float)ai.y;
      v2f b = *(const v2f*)(bp + k);
      acc = wmma_f32(a, b, acc);
    }
    float* op = den + (size_t)it * 16 * 16 + r;
#pragma unroll
    for (int vr = 0; vr < 8; ++vr) op[(vr + 8 * half) * 16] = acc[vr];
  }
}

// ---------------------------------------------------------------------------
// K4: out[n, h*64+f] /= den[n, h]
// ---------------------------------------------------------------------------
__global__ void gat_k4_div(float* __restrict__ out, const float* __restrict__ den) {
  const int id = blockIdx.x * 256 + threadIdx.x;  // 0..N*OUTC-1
  const int n  = id >> 9;
  const int hd = (id >> 6) & 7;
  out[id] /= den[n * 16 + hd];
}

// ---------------------------------------------------------------------------
extern "C" void kernel_launch(void* const* d_in, const int* in_sizes, int n_in,
                              void* d_out, int out_size, void* d_ws, size_t ws_size,
                              hipStream_t stream) {
  const float* h   = (const float*)d_in[0];   // 4096x512
  const int*   adj = (const int*)  d_in[1];   // 4096x4096x1
  const float* w   = (const float*)d_in[2];   // 512x512
  const float* aw  = (const float*)d_in[3];   // 1x128
  float* out = (float*)d_out;                 // 4096x512

  char* ws = (char*)d_ws;
  size_t off = 0;
  float* g     = (float*)(ws + off); off += (size_t)N_NODES * OUTC * 4;     // 8 MB
  float* bt    = (float*)(ws + off); off += (size_t)OUTC * N_NODES * 4;     // 8 MB
  float* ej    = (float*)(ws + off); off += (size_t)N_NODES * N_HEADS * 4;  // 128 KB
  float* M     = (float*)(ws + off); off += 256;                            // 8 floats
  float* eextT = (float*)(ws + off); off += (size_t)16 * N_NODES * 4;       // 256 KB
  float* den   = (float*)(ws + off); off += (size_t)N_NODES * 16 * 4;       // 256 KB
  (void)ws_size; (void)in_sizes; (void)n_in; (void)out_size;

  gat_k1_gemm      <<<256,  256, 0, stream>>>(h, w, g);
  gat_k2a_ej       <<<128,  256, 0, stream>>>(g, aw, ej);
  gat_k2b_max      <<<1,    256, 0, stream>>>(ej, M);
  gat_k2t_transpose<<<dim3(16, 128), 256, 0, stream>>>(g, ej, M, bt);
  gat_k2e_eext     <<<256,  256, 0, stream>>>(ej, M, eextT);
  gat_k3_attn      <<<288,  256, 0, stream>>>(adj, bt, eextT, out, den);
  gat_k4_div       <<<8192, 256, 0, stream>>>(out, den);
}